// OhemCrossEntropy2d_42417097016564
// MI455X (gfx1250) — compile-verified
//
#include <hip/hip_runtime.h>
#include <hip/hip_bf16.h>
#include <stdint.h>

#ifndef __has_builtin
#define __has_builtin(x) 0
#endif

// ---- CDNA5 async global->LDS path (guarded; falls back to direct loads) ----
#if __has_builtin(__builtin_amdgcn_global_load_async_to_lds_b128)
#define USE_ASYNC_LDS 1
#else
#define USE_ASYNC_LDS 0
#endif

#if __has_builtin(__builtin_amdgcn_s_wait_asynccnt)
#define WAIT_ASYNC(n) __builtin_amdgcn_s_wait_asynccnt(n)
#else
#define WAIT_ASYNC(n) asm volatile("s_wait_asynccnt %0" ::"i"(n) : "memory")
#endif

// Problem constants
#define NN 8
#define CC 19
#define HH 512
#define WW 1024
#define HW (HH * WW)      // 524288 = 2^19
#define PP (NN * HW)      // 4194304
#define THRESH_F 0.7f
#define MIN_KEPT_K 100000
#define IGNORE_L 255

struct Ctrl {
  unsigned num_valid;  // [0]
  unsigned prefix;     // [1] radix-select prefix
  unsigned rank;       // [2] remaining rank within prefix
  unsigned do_ohem;    // [3]
  float    threshold;  // [4]
  float    loss_sum;   // [5]
  unsigned kept_cnt;   // [6]
  unsigned pad;        // [7]
};

// ---------------------------------------------------------------------------
__global__ void init_ws(unsigned* __restrict__ hist, unsigned* __restrict__ ctrl_u) {
  int i = blockIdx.x * blockDim.x + threadIdx.x;
  if (i < 4096) hist[i] = 0u;
  if (i < 8) ctrl_u[i] = 0u;
}

// ---------------------------------------------------------------------------
// Pass 1: per-pixel log-softmax; store pred (inf if ignored) and -true_logp.
__global__ void main_pass(const float* __restrict__ predict,
                          const int* __restrict__ target,
                          float* __restrict__ pred_out,
                          float* __restrict__ nlp_out,
                          unsigned* __restrict__ num_valid_g) {
  const int p = blockIdx.x * blockDim.x + threadIdx.x;  // grid covers PP exactly
  const int n  = p >> 19;          // p / HW
  const int hw = p & (HW - 1);     // p % HW
  const float* base = predict + (size_t)n * CC * HW + hw;

  const int lbl = target[p];
  const bool valid = (lbl != IGNORE_L);
  const int sl = valid ? lbl : 0;

  float v[CC];
  float m = -3.402823466e38f;
  float chosen = 0.0f;
#pragma unroll
  for (int c = 0; c < CC; ++c) {
    float x = base[(size_t)c * HW];
    v[c] = x;
    m = fmaxf(m, x);
    chosen = (c == sl) ? x : chosen;   // avoids dynamic register indexing
  }
  float s = 0.0f;
#pragma unroll
  for (int c = 0; c < CC; ++c) s += __expf(v[c] - m);
  const float logZ = m + __logf(s);

  pred_out[p] = valid ? __expf(chosen - logZ) : __builtin_inff();
  nlp_out[p]  = valid ? (logZ - chosen) : 0.0f;   // -true_logp

  __shared__ unsigned blkv;
  if (threadIdx.x == 0) blkv = 0u;
  __syncthreads();
  unsigned long long b = __ballot(valid ? 1 : 0);
  if ((threadIdx.x & 31) == 0) atomicAdd(&blkv, (unsigned)__popcll(b));
  __syncthreads();
  if (threadIdx.x == 0) atomicAdd(num_valid_g, blkv);
}

// ---------------------------------------------------------------------------
__device__ __forceinline__ void hist_elem(float x, unsigned* lh, unsigned prefix, int round) {
  const unsigned bits = __float_as_uint(x);   // positive floats: bit order == value order
  if (round == 0) {
    atomicAdd(&lh[bits >> 20], 1u);           // top 12 bits
  } else if (round == 1) {
    if ((bits >> 20) == prefix) atomicAdd(&lh[(bits >> 8) & 0xFFFu], 1u);  // mid 12 bits
  } else {
    if ((bits >> 8) == prefix) atomicAdd(&lh[bits & 0xFFu], 1u);           // low 8 bits
  }
}

#if USE_ASYNC_LDS
// Builtin prototype (from diagnostic): v4i AS1* src, v4i AS3* dst, imm offset, imm cpol
typedef int vint4_t __attribute__((vector_size(16)));
typedef __attribute__((address_space(1))) vint4_t* gv4_ptr;
typedef __attribute__((address_space(3))) vint4_t* lv4_ptr;

__device__ __forceinline__ void issue_async_tile(const float* gsrc, float* lds_dst) {
  __builtin_amdgcn_global_load_async_to_lds_b128(
      (gv4_ptr)gsrc, (lv4_ptr)lds_dst, /*offset=*/0, /*cpol=*/0);
}
#endif

// Histogram round over pred[] (16.8 MB stream). Async double-buffered LDS staging.
__global__ void hist_pass(const float* __restrict__ pred,
                          unsigned* __restrict__ hist,
                          const Ctrl* __restrict__ ctrl,
                          int round) {
  __shared__ unsigned lh[4096];
  for (int i = threadIdx.x; i < 4096; i += blockDim.x) lh[i] = 0u;
  __syncthreads();

  const unsigned prefix = ctrl->prefix;
  const int tid = threadIdx.x;
  const int ntiles = PP / 1024;  // 1024 floats per tile: one b128 per lane

#if USE_ASYNC_LDS
  __shared__ float stage[2][1024];
  int t = blockIdx.x;
  if (t < ntiles) issue_async_tile(pred + (size_t)t * 1024 + tid * 4, &stage[0][tid * 4]);
  int buf = 0;
  for (; t < ntiles; t += gridDim.x, buf ^= 1) {
    const int nxt = t + (int)gridDim.x;
    if (nxt < ntiles) {
      issue_async_tile(pred + (size_t)nxt * 1024 + tid * 4, &stage[buf ^ 1][tid * 4]);
      WAIT_ASYNC(1);   // current tile complete (async loads complete in order)
    } else {
      WAIT_ASYNC(0);
    }
    // each lane reads only its own staged slot -> no barrier needed
    float4 x = ((const float4*)&stage[buf][0])[tid];
    hist_elem(x.x, lh, prefix, round);
    hist_elem(x.y, lh, prefix, round);
    hist_elem(x.z, lh, prefix, round);
    hist_elem(x.w, lh, prefix, round);
  }
#else
  const float4* p4 = (const float4*)pred;
  const int stride = gridDim.x * blockDim.x;
  for (int i = blockIdx.x * blockDim.x + tid; i < PP / 4; i += stride) {
    __builtin_prefetch(&p4[i + stride], 0, 0);  // global_prefetch_b8
    float4 x = p4[i];
    hist_elem(x.x, lh, prefix, round);
    hist_elem(x.y, lh, prefix, round);
    hist_elem(x.z, lh, prefix, round);
    hist_elem(x.w, lh, prefix, round);
  }
#endif

  __syncthreads();
  for (int i = threadIdx.x; i < 4096; i += blockDim.x) {
    unsigned v = lh[i];
    if (v) atomicAdd(&hist[i], v);
  }
}

// ---------------------------------------------------------------------------
// Single-block scan+select: locate bin holding the remaining rank, update ctrl,
// and zero the histogram for the next round.
__global__ void select_pass(unsigned* __restrict__ hist, Ctrl* __restrict__ ctrl, int round) {
  const int nbins = (round == 2) ? 256 : 4096;
  const int per = nbins / 256;  // 16 or 1
  const int t = threadIdx.x;

  __shared__ unsigned base_[256];
  __shared__ unsigned ksh;

  unsigned vals[16];
  unsigned local = 0;
  for (int j = 0; j < per; ++j) {
    vals[j] = hist[t * per + j];
    local += vals[j];
  }
  base_[t] = local;
  __syncthreads();

  if (t == 0) {
    unsigned acc = 0;
    for (int i = 0; i < 256; ++i) {  // serial exclusive scan of 256 partials
      unsigned v = base_[i];
      base_[i] = acc;
      acc += v;
    }
    unsigned k;
    if (round == 0) {
      unsigned nv = ctrl->num_valid;
      unsigned kk = (nv < (unsigned)MIN_KEPT_K) ? nv : (unsigned)MIN_KEPT_K;
      k = (kk > 0u) ? (kk - 1u) : 0u;  // kth_idx
    } else {
      k = ctrl->rank;
    }
    ksh = k;
  }
  __syncthreads();

  const unsigned k = ksh;
  if (local > 0u && k >= base_[t] && k < base_[t] + local) {
    unsigned rem = k - base_[t];
    int bsel = 0;
    for (int j = 0; j < per; ++j) {
      if (rem < vals[j]) { bsel = j; break; }
      rem -= vals[j];
    }
    const unsigned bin = (unsigned)(t * per + bsel);
    if (round == 0) {
      ctrl->prefix = bin;
      ctrl->rank = rem;
    } else if (round == 1) {
      ctrl->prefix = (ctrl->prefix << 12) | bin;
      ctrl->rank = rem;
    } else {
      const unsigned bits = (ctrl->prefix << 8) | bin;
      const float thr = __uint_as_float(bits);  // exact k-th smallest pred
      ctrl->threshold = (thr > THRESH_F) ? thr : THRESH_F;
      ctrl->do_ohem = (ctrl->num_valid > (unsigned)MIN_KEPT_K) ? 1u : 0u;
      ctrl->loss_sum = 0.0f;
      ctrl->kept_cnt = 0u;
    }
  }
  for (int j = 0; j < per; ++j) hist[t * per + j] = 0u;  // reset for next round
}

// ---------------------------------------------------------------------------
__global__ void reduce_pass(const float* __restrict__ pred,
                            const float* __restrict__ nlp,
                            Ctrl* __restrict__ ctrl) {
  const float threshold = ctrl->threshold;
  const bool do_ohem = (ctrl->do_ohem != 0u);
  const int stride = gridDim.x * blockDim.x;

  float s = 0.0f;
  unsigned c = 0u;
  for (int i = blockIdx.x * blockDim.x + threadIdx.x; i < PP; i += stride) {
    __builtin_prefetch(&pred[i + stride], 0, 0);  // global_prefetch_b8
    const float p = pred[i];
    const bool valid = (p < __builtin_inff());
    const bool kept = do_ohem ? (valid && p <= threshold) : valid;
    if (kept) { s += nlp[i]; c += 1u; }
  }

  __shared__ float ss[256];
  __shared__ unsigned sc[256];
  ss[threadIdx.x] = s;
  sc[threadIdx.x] = c;
  __syncthreads();
  for (int o = 128; o > 0; o >>= 1) {
    if ((int)threadIdx.x < o) {
      ss[threadIdx.x] += ss[threadIdx.x + o];
      sc[threadIdx.x] += sc[threadIdx.x + o];
    }
    __syncthreads();
  }
  if (threadIdx.x == 0) {
    atomicAdd(&ctrl->loss_sum, ss[0]);
    atomicAdd(&ctrl->kept_cnt, sc[0]);
  }
}

__global__ void finalize_k(const Ctrl* __restrict__ ctrl, float* __restrict__ out) {
  const unsigned c = ctrl->kept_cnt;
  out[0] = ctrl->loss_sum / (float)(c > 0u ? c : 1u);
}

// ---------------------------------------------------------------------------
extern "C" void kernel_launch(void* const* d_in, const int* in_sizes, int n_in,
                              void* d_out, int out_size, void* d_ws, size_t ws_size,
                              hipStream_t stream) {
  (void)in_sizes; (void)n_in; (void)out_size; (void)ws_size;
  const float* predict = (const float*)d_in[0];
  const int* target = (const int*)d_in[1];
  float* out = (float*)d_out;

  float* pred = (float*)d_ws;                 // PP floats
  float* nlp = pred + PP;                     // PP floats
  unsigned* hist = (unsigned*)(nlp + PP);     // 4096 u32
  Ctrl* ctrl = (Ctrl*)(hist + 4096);          // 8 u32

  init_ws<<<16, 256, 0, stream>>>(hist, (unsigned*)ctrl);
  main_pass<<<PP / 256, 256, 0, stream>>>(predict, target, pred, nlp, (unsigned*)ctrl);
  hist_pass<<<512, 256, 0, stream>>>(pred, hist, ctrl, 0);
  select_pass<<<1, 256, 0, stream>>>(hist, ctrl, 0);
  hist_pass<<<512, 256, 0, stream>>>(pred, hist, ctrl, 1);
  select_pass<<<1, 256, 0, stream>>>(hist, ctrl, 1);
  hist_pass<<<512, 256, 0, stream>>>(pred, hist, ctrl, 2);
  select_pass<<<1, 256, 0, stream>>>(hist, ctrl, 2);
  reduce_pass<<<1024, 256, 0, stream>>>(pred, nlp, ctrl);
  finalize_k<<<1, 1, 0, stream>>>(ctrl, out);
}